// GDMLPredict_5437428597140
// MI455X (gfx1250) — compile-verified
//
#include <hip/hip_runtime.h>
#include <math.h>

// ---------------- problem constants ----------------
constexpr int NA = 21;          // atoms
constexpr int DD = 210;         // pairs = NA*(NA-1)/2
constexpr int BB = 64;          // batch
constexpr int MM = 6000;        // training points
constexpr int XX = 63;          // 3*NA
constexpr int DP = 224;         // D padded (mult of 16)
constexpr int XP = 64;          // X padded
constexpr int NCH = 8;          // m-chunks for hessian partials
constexpr int MTILES = MM / 16; // 375
constexpr int TPC = (MTILES + NCH - 1) / NCH; // 47
constexpr int ROWQW = DD / 2;   // 105 b64 chunks per 210-float row
constexpr int TILEQW = 16 * ROWQW; // 1680 b64 chunks per 16-row tile

constexpr float QC    = 0.22360679774997896f;   // sqrt(5)/SIG, SIG=10
constexpr float Q4_3  = 0.0025f / 3.0f;          // q^4/3
constexpr float INVQ2 = 20.0f;                   // 1/q^2
constexpr float STDC  = 1.0f;
constexpr float CCONST= 0.0f;

// ---------------- workspace layout (floats) ----------------
constexpr size_t OFF_XS   = 0;
constexpr size_t OFF_DKL  = OFF_XS  + (size_t)BB*DD;        // dkl: B*D*3
constexpr size_t OFF_JAC  = OFF_DKL + (size_t)BB*DD*3;      // jac dense: B*DP*XP
constexpr size_t OFF_W1   = OFF_JAC + (size_t)BB*DP*XP;
constexpr size_t OFF_E    = OFF_W1  + (size_t)BB*MM;
constexpr size_t OFF_XA   = OFF_E   + (size_t)BB*MM;
constexpr size_t OFF_CA   = OFF_XA  + (size_t)BB*MM;
constexpr size_t OFF_SEXA = OFF_CA  + (size_t)BB*MM;
constexpr size_t OFF_GX   = OFF_SEXA+ (size_t)BB;
constexpr size_t OFF_P    = OFF_GX  + (size_t)BB*DP;
constexpr size_t OFF_Q    = OFF_P   + (size_t)NCH*BB*4096;

// ---------------- WMMA helper (f32 16x16x4, full f32 precision) ----------------
typedef float v2f __attribute__((ext_vector_type(2)));
typedef float v8f __attribute__((ext_vector_type(8)));

__device__ __forceinline__ v8f wmma4(v2f a, v2f b, v8f c) {
    return __builtin_amdgcn_wmma_f32_16x16x4_f32(
        false, a, false, b, (short)0, c, false, false);
}

// async global -> LDS copy of one 64-bit chunk (ASYNCcnt tracked)
__device__ __forceinline__ void async_ld_b64(const float* gp, const float* lp) {
    const unsigned ldsb = (unsigned)(uintptr_t)lp;   // flat low 32 bits == LDS byte addr
    asm volatile("global_load_async_to_lds_b64 %0, %1, off"
                 :: "v"(ldsb), "v"(gp) : "memory");
}
__device__ __forceinline__ void wait_async0() {
    asm volatile("s_wait_asynccnt 0x0" ::: "memory");
}

__device__ __forceinline__ void pair_from_d(int d, int& i, int& j) {
    int ii = (int)((1.0f + sqrtf(1.0f + 8.0f * (float)d)) * 0.5f);
    while (ii * (ii - 1) / 2 > d)  --ii;
    while ((ii + 1) * ii / 2 <= d) ++ii;
    i = ii;
    j = d - ii * (ii - 1) / 2;
}

// ---------------- kernel 1: per-batch geometry: xs, dkl, dense jac ----------------
__global__ __launch_bounds__(256) void k_prep(const float* __restrict__ r,
                                              float* __restrict__ xs_ws,
                                              float* __restrict__ dkl_ws,
                                              float* __restrict__ jac_ws) {
    const int b = blockIdx.x, tid = threadIdx.x;
    __shared__ float s_r[3 * NA];
    if (tid < 3 * NA) s_r[tid] = r[b * 3 * NA + tid];
    float* jb = jac_ws + (size_t)b * DP * XP;
    for (int idx = tid; idx < DP * XP; idx += 256) jb[idx] = 0.0f;
    __syncthreads();
    if (tid < DD) {
        const int d = tid;
        int i, j; pair_from_d(d, i, j);
        const float dx = s_r[3*i+0] - s_r[3*j+0];
        const float dy = s_r[3*i+1] - s_r[3*j+1];
        const float dz = s_r[3*i+2] - s_r[3*j+2];
        const float inv = 1.0f / sqrtf(dx*dx + dy*dy + dz*dz);
        xs_ws[b * DD + d] = inv;
        dkl_ws[((size_t)b * DD + d) * 3 + 0] = dx;
        dkl_ws[((size_t)b * DD + d) * 3 + 1] = dy;
        dkl_ws[((size_t)b * DD + d) * 3 + 2] = dz;
        const float x3 = inv * inv * inv;
        const float c0 = -x3 * dx, c1 = -x3 * dy, c2 = -x3 * dz;
        jb[d * XP + 3*i + 0] =  c0; jb[d * XP + 3*i + 1] =  c1; jb[d * XP + 3*i + 2] =  c2;
        jb[d * XP + 3*j + 0] = -c0; jb[d * XP + 3*j + 1] = -c1; jb[d * XP + 3*j + 2] = -c2;
    }
}

// ---------------- kernel 2: per-(b,m) scalars (wave32 per pair) ----------------
__global__ __launch_bounds__(128) void k_scalars(const float* __restrict__ xtr,
                                                 const float* __restrict__ alph,
                                                 const float* __restrict__ xs_ws,
                                                 float* __restrict__ w1_ws,
                                                 float* __restrict__ e_ws,
                                                 float* __restrict__ xa_ws,
                                                 float* __restrict__ ca_ws) {
    const int gw   = blockIdx.x * 4 + (threadIdx.x >> 5);
    const int lane = threadIdx.x & 31;
    const int b = gw / MM, m = gw % MM;
    const float* xr  = xtr  + (size_t)m * DD;
    const float* ar  = alph + (size_t)m * DD;
    const float* xsb = xs_ws + (size_t)b * DD;
    float d2 = 0.0f, xa = 0.0f;
    for (int d = lane; d < DD; d += 32) {
        const float diff = xsb[d] - xr[d];
        d2 += diff * diff;
        xa += diff * ar[d];
    }
    for (int off = 16; off > 0; off >>= 1) {
        d2 += __shfl_down(d2, off);
        xa += __shfl_down(xa, off);
    }
    if (lane == 0) {
        const float xd = sqrtf(d2);
        const float e  = Q4_3 * expf(-QC * xd);
        w1_ws[gw] = e * (1.0f + QC * xd) * INVQ2;
        e_ws[gw]  = e;
        xa_ws[gw] = xa;
        ca_ws[gw] = e * xa * QC / xd;
    }
}

// ---------------- kernel 3: energy + Sexa reduction ----------------
__global__ __launch_bounds__(256) void k_energy(const float* __restrict__ w1_ws,
                                                const float* __restrict__ e_ws,
                                                const float* __restrict__ xa_ws,
                                                float* __restrict__ energy_out,
                                                float* __restrict__ sexa_ws) {
    const int b = blockIdx.x, tid = threadIdx.x;
    __shared__ float s1[256], s2[256];
    float a1 = 0.0f, a2 = 0.0f;
    for (int m = tid; m < MM; m += 256) {
        const float xa = xa_ws[(size_t)b * MM + m];
        a1 += w1_ws[(size_t)b * MM + m] * xa;
        a2 += e_ws [(size_t)b * MM + m] * xa;
    }
    s1[tid] = a1; s2[tid] = a2;
    __syncthreads();
    for (int s = 128; s > 0; s >>= 1) {
        if (tid < s) { s1[tid] += s1[tid + s]; s2[tid] += s2[tid + s]; }
        __syncthreads();
    }
    if (tid == 0) {
        energy_out[b] = s1[0] * STDC + CCONST;
        sexa_ws[b]    = s2[0];
    }
}

// ---------------- kernel 4: grad_x (B x D) ----------------
__global__ __launch_bounds__(256) void k_gradx(const float* __restrict__ xtr,
                                               const float* __restrict__ alph,
                                               const float* __restrict__ w1_ws,
                                               const float* __restrict__ e_ws,
                                               const float* __restrict__ xa_ws,
                                               const float* __restrict__ xs_ws,
                                               const float* __restrict__ sexa_ws,
                                               float* __restrict__ gx_ws) {
    const int b = blockIdx.x, tid = threadIdx.x;
    __shared__ float sw[200], se[200];
    float a1 = 0.0f, a2 = 0.0f;
    for (int mc = 0; mc < MM; mc += 200) {
        if (tid < 200) {
            const int m = mc + tid;
            sw[tid] = w1_ws[(size_t)b * MM + m];
            se[tid] = e_ws[(size_t)b * MM + m] * xa_ws[(size_t)b * MM + m];
        }
        __syncthreads();
        if (tid < DD) {
            for (int mm = 0; mm < 200; ++mm) {
                const int m = mc + mm;
                a1 += sw[mm] * alph[(size_t)m * DD + tid];
                a2 += se[mm] * xtr [(size_t)m * DD + tid];
            }
        }
        __syncthreads();
    }
    if (tid < DP) {
        float g = 0.0f;
        if (tid < DD) g = a1 + a2 - xs_ws[(size_t)b * DD + tid] * sexa_ws[b];
        gx_ws[(size_t)b * DP + tid] = g;
    }
}

// ---------------- kernel 5: grad = grad_x @ jac ----------------
__global__ __launch_bounds__(64) void k_grad(const float* __restrict__ gx_ws,
                                             const float* __restrict__ jac_ws,
                                             float* __restrict__ out_grad) {
    const int b = blockIdx.x, x = threadIdx.x;
    const float* jb = jac_ws + (size_t)b * DP * XP;
    const float* g  = gx_ws + (size_t)b * DP;
    float acc = 0.0f;
    for (int d = 0; d < DD; ++d) acc += g[d] * jb[d * XP + x];
    if (x < XX) out_grad[(size_t)b * XX + x] = acc * STDC;
}

// ---------------- kernel 6: fused Hessian partials ----------------
// Per workgroup (b, chunk), double-buffered async global->LDS staging:
//   stage1 (WMMA, K=224): G = Xtr_tile @ jac ; AJ = Alph_tile @ jac ;
//                          XJ = sv - G  with sv = xs^T @ jac (once per block)
//   stage2 (WMMA, K=16):  P += XJ^T diag(ca) XJ ;  Q += AJ^T diag(e) XJ
__global__ __launch_bounds__(128) void k_hess_partial(const float* __restrict__ xtr,
                                                      const float* __restrict__ alph,
                                                      const float* __restrict__ xs_ws,
                                                      const float* __restrict__ jac_ws,
                                                      const float* __restrict__ ca_ws,
                                                      const float* __restrict__ e_ws,
                                                      float* __restrict__ Pws,
                                                      float* __restrict__ Qws) {
    __shared__ float s_jac[DP * XP];        // 56 KB
    __shared__ float s_tr[2][16 * DP];      // raw xs_train tiles (double buffer), 28 KB
    __shared__ float s_ah[2][16 * DP];      // raw alpha tiles (double buffer), 28 KB
    __shared__ float s_xs[DP];
    __shared__ float s_sv[XP];              // sv = xs^T @ jac
    __shared__ float s_xj[16 * XP];
    __shared__ float s_aj[16 * XP];
    __shared__ float s_ca[16];
    __shared__ float s_e[16];

    const int b     = blockIdx.x / NCH;
    const int chunk = blockIdx.x % NCH;
    const int tid  = threadIdx.x;
    const int wave = tid >> 5, lane = tid & 31;
    const int hl = lane >> 4, ln = lane & 15;

    const int t0 = chunk * TPC;
    const int t1 = (t0 + TPC < MTILES) ? (t0 + TPC) : MTILES;
    const int ncol = wave * 16 + ln;

    // kick off async staging of the first m-tile while jac loads
    {
        const int mbase = t0 * 16;
        for (int idx = tid; idx < 2 * TILEQW; idx += 128) {
            const int sel = (idx >= TILEQW) ? 1 : 0;
            const int r   = idx - sel * TILEQW;
            const int row = r / ROWQW;
            const int c2  = (r - row * ROWQW) * 2;
            const float* gp = (sel ? alph : xtr) + (size_t)(mbase + row) * DD + c2;
            const float* lp = (sel ? &s_ah[0][0] : &s_tr[0][0]) + row * DP + c2;
            async_ld_b64(gp, lp);
        }
    }
    for (int idx = tid; idx < DP * XP; idx += 128)
        s_jac[idx] = jac_ws[(size_t)b * DP * XP + idx];
    for (int idx = tid; idx < DP; idx += 128)
        s_xs[idx] = (idx < DD) ? xs_ws[(size_t)b * DD + idx] : 0.0f;
    __syncthreads();
    if (tid < XP) {
        float acc = 0.0f;
        for (int d = 0; d < DD; ++d) acc += s_xs[d] * s_jac[d * XP + tid];
        s_sv[tid] = acc;
    }

    v8f accP[4], accQ[4];
    for (int t = 0; t < 4; ++t) { accP[t] = v8f{}; accQ[t] = v8f{}; }

    for (int mt = t0; mt < t1; ++mt) {
        const int cur = (mt - t0) & 1, nxt = cur ^ 1;
        const int mbase = mt * 16;

        wait_async0();        // this wave's async loads for buf[cur] complete
        __syncthreads();      // all waves' loads visible; prev stage2 done

        if (tid < 16) {
            s_ca[tid] = ca_ws[(size_t)b * MM + mbase + tid];
            s_e[tid]  = e_ws [(size_t)b * MM + mbase + tid];
        }
        // prefetch next tile into buf[nxt], overlapped with WMMA below
        {
            const int pmt = (mt + 1 < MTILES) ? (mt + 1) : (MTILES - 1);
            const int pb  = pmt * 16;
            for (int idx = tid; idx < 2 * TILEQW; idx += 128) {
                const int sel = (idx >= TILEQW) ? 1 : 0;
                const int r   = idx - sel * TILEQW;
                const int row = r / ROWQW;
                const int c2  = (r - row * ROWQW) * 2;
                const float* gp = (sel ? alph : xtr) + (size_t)(pb + row) * DD + c2;
                const float* lp = (sel ? &s_ah[nxt][0] : &s_tr[nxt][0]) + row * DP + c2;
                async_ld_b64(gp, lp);
            }
        }

        // stage 1: G / AJ tiles (each wave owns a 16-col strip); K pads are
        // harmless because jac's padded K-rows are zero.
        v8f cx = v8f{}, ca8 = v8f{};
        for (int k = 0; k < DP; k += 4) {
            const int k0 = k + 2 * hl, k1 = k0 + 1;
            v2f bb; bb.x = s_jac[k0 * XP + ncol];     bb.y = s_jac[k1 * XP + ncol];
            v2f ax; ax.x = s_tr[cur][ln * DP + k0];   ax.y = s_tr[cur][ln * DP + k1];
            v2f aa; aa.x = s_ah[cur][ln * DP + k0];   aa.y = s_ah[cur][ln * DP + k1];
            cx  = wmma4(ax, bb, cx);
            ca8 = wmma4(aa, bb, ca8);
        }
        for (int rr = 0; rr < 8; ++rr) {
            const int row = rr + 8 * hl;
            s_xj[row * XP + ncol] = s_sv[ncol] - cx[rr];   // XJ = sv - Xtr@jac
            s_aj[row * XP + ncol] = ca8[rr];
        }
        __syncthreads();

        // stage 2: rank-16 updates into P and Q (each wave owns a 16-row strip)
        const int rs = wave * 16;
        for (int nt = 0; nt < 4; ++nt) {
            v8f p = accP[nt], q = accQ[nt];
            for (int k = 0; k < 16; k += 4) {
                const int k0 = k + 2 * hl, k1 = k0 + 1;
                v2f aP; aP.x = s_xj[k0 * XP + rs + ln]; aP.y = s_xj[k1 * XP + rs + ln];
                v2f aQ; aQ.x = s_aj[k0 * XP + rs + ln]; aQ.y = s_aj[k1 * XP + rs + ln];
                const float x0 = s_xj[k0 * XP + nt * 16 + ln];
                const float x1 = s_xj[k1 * XP + nt * 16 + ln];
                v2f bP; bP.x = s_ca[k0] * x0; bP.y = s_ca[k1] * x1;
                v2f bQ; bQ.x = s_e [k0] * x0; bQ.y = s_e [k1] * x1;
                p = wmma4(aP, bP, p);
                q = wmma4(aQ, bQ, q);
            }
            accP[nt] = p; accQ[nt] = q;
        }
    }
    wait_async0();   // drain trailing prefetch before exit

    float* Pp = Pws + ((size_t)(chunk * BB + b)) * 4096;
    float* Qp = Qws + ((size_t)(chunk * BB + b)) * 4096;
    for (int nt = 0; nt < 4; ++nt)
        for (int rr = 0; rr < 8; ++rr) {
            const int row = wave * 16 + rr + 8 * hl;
            const int col = nt * 16 + ln;
            Pp[row * 64 + col] = accP[nt][rr];
            Qp[row * 64 + col] = accQ[nt][rr];
        }
}

// ---------------- kernel 7: final Hessian assembly ----------------
__global__ __launch_bounds__(128) void k_hess_final(const float* __restrict__ Pws,
                                                    const float* __restrict__ Qws,
                                                    const float* __restrict__ jac_ws,
                                                    const float* __restrict__ sexa_ws,
                                                    const float* __restrict__ gx_ws,
                                                    const float* __restrict__ xs_ws,
                                                    const float* __restrict__ dkl_ws,
                                                    float* __restrict__ out_hess) {
    __shared__ float s_jac[DP * XP];
    __shared__ float s_h[64 * 64];
    const int b = blockIdx.x, tid = threadIdx.x;

    // sum partial P - Q - Q^T over chunks
    for (int idx = tid; idx < 4096; idx += 128) {
        const int row = idx >> 6, col = idx & 63;
        float acc = 0.0f;
        for (int c = 0; c < NCH; ++c) {
            const float* Pp = Pws + ((size_t)(c * BB + b)) * 4096;
            const float* Qp = Qws + ((size_t)(c * BB + b)) * 4096;
            acc += Pp[idx] - Qp[idx] - Qp[col * 64 + row];
        }
        s_h[idx] = acc;
    }
    for (int idx = tid; idx < DP * XP; idx += 128)
        s_jac[idx] = jac_ws[(size_t)b * DP * XP + idx];
    __syncthreads();

    // subtract Sexa * (jac^T jac) via WMMA (K = 224)
    const int wave = tid >> 5, lane = tid & 31;
    const int hl = lane >> 4, ln = lane & 15;
    const int rs = wave * 16;
    const float sexa = sexa_ws[b];
    for (int nt = 0; nt < 4; ++nt) {
        v8f c = v8f{};
        for (int k = 0; k < DP; k += 4) {
            const int k0 = k + 2 * hl, k1 = k0 + 1;
            v2f a;  a.x  = s_jac[k0 * XP + rs + ln];      a.y  = s_jac[k1 * XP + rs + ln];
            v2f bb; bb.x = s_jac[k0 * XP + nt * 16 + ln]; bb.y = s_jac[k1 * XP + nt * 16 + ln];
            c = wmma4(a, bb, c);
        }
        for (int rr = 0; rr < 8; ++rr) {
            const int row = rs + rr + 8 * hl;
            const int col = nt * 16 + ln;
            s_h[row * 64 + col] -= sexa * c[rr];
        }
    }
    __syncthreads();

    // h_blk sparse scatter: 9 threads each own a (c,cc) offset (conflict-free)
    if (tid < 9) {
        const int c = tid / 3, cc = tid % 3;
        for (int d = 0; d < DD; ++d) {
            int i, j; pair_from_d(d, i, j);
            const float x  = xs_ws[(size_t)b * DD + d];
            const float gx = gx_ws[(size_t)b * DP + d];
            const float x3 = x * x * x, x5 = x3 * x * x;
            const float dc  = dkl_ws[((size_t)b * DD + d) * 3 + c];
            const float dcc = dkl_ws[((size_t)b * DD + d) * 3 + cc];
            float h = 3.0f * gx * x5 * dc * dcc;
            if (c == cc) h -= gx * x3;
            s_h[(3*i + c) * 64 + (3*j + cc)] -= h;
            s_h[(3*j + c) * 64 + (3*i + cc)] -= h;
            s_h[(3*i + c) * 64 + (3*i + cc)] += h;
            s_h[(3*j + c) * 64 + (3*j + cc)] += h;
        }
    }
    __syncthreads();

    float* oh = out_hess + (size_t)b * XX * XX;
    for (int idx = tid; idx < XX * XX; idx += 128) {
        const int row = idx / XX, col = idx - row * XX;
        oh[idx] = s_h[row * 64 + col] * STDC;
    }
}

// ---------------- launcher ----------------
extern "C" void kernel_launch(void* const* d_in, const int* in_sizes, int n_in,
                              void* d_out, int out_size, void* d_ws, size_t ws_size,
                              hipStream_t stream) {
    (void)in_sizes; (void)n_in; (void)out_size; (void)ws_size;
    const float* r    = (const float*)d_in[0];
    const float* xtr  = (const float*)d_in[1];
    const float* alph = (const float*)d_in[2];
    float* out = (float*)d_out;
    float* ws  = (float*)d_ws;

    float* xs_ws   = ws + OFF_XS;
    float* dkl_ws  = ws + OFF_DKL;
    float* jac_ws  = ws + OFF_JAC;
    float* w1_ws   = ws + OFF_W1;
    float* e_ws    = ws + OFF_E;
    float* xa_ws   = ws + OFF_XA;
    float* ca_ws   = ws + OFF_CA;
    float* sexa_ws = ws + OFF_SEXA;
    float* gx_ws   = ws + OFF_GX;
    float* Pws     = ws + OFF_P;
    float* Qws     = ws + OFF_Q;

    float* out_energy = out;                 // B
    float* out_grad   = out + BB;            // B*63
    float* out_hess   = out + BB + BB * XX;  // B*63*63

    k_prep        <<<BB,            256, 0, stream>>>(r, xs_ws, dkl_ws, jac_ws);
    k_scalars     <<<(BB * MM) / 4, 128, 0, stream>>>(xtr, alph, xs_ws, w1_ws, e_ws, xa_ws, ca_ws);
    k_energy      <<<BB,            256, 0, stream>>>(w1_ws, e_ws, xa_ws, out_energy, sexa_ws);
    k_gradx       <<<BB,            256, 0, stream>>>(xtr, alph, w1_ws, e_ws, xa_ws, xs_ws, sexa_ws, gx_ws);
    k_grad        <<<BB,             64, 0, stream>>>(gx_ws, jac_ws, out_grad);
    k_hess_partial<<<BB * NCH,      128, 0, stream>>>(xtr, alph, xs_ws, jac_ws, ca_ws, e_ws, Pws, Qws);
    k_hess_final  <<<BB,            128, 0, stream>>>(Pws, Qws, jac_ws, sexa_ws, gx_ws, xs_ws, dkl_ws, out_hess);
}